// CausalAttention_83743272337989
// MI455X (gfx1250) — compile-verified
//
#include <hip/hip_runtime.h>
#include <hip/hip_bf16.h>

typedef __attribute__((ext_vector_type(2))) float v2f;
typedef __attribute__((ext_vector_type(8))) float v8f;

#define WMMA_F32(a, b, c) \
  __builtin_amdgcn_wmma_f32_16x16x4_f32(false, (a), false, (b), (short)0, (c), false, false)

namespace {
constexpr int SEQ   = 2048;
constexpr int DM    = 1024;   // d_model
constexpr int NH    = 16;
constexpr int DH    = 64;     // d_head
constexpr int BATCH = 2;
constexpr int MROWS = BATCH * SEQ;  // 4096
}

// ---------------------------------------------------------------------------
// C[M, 1024] = A[M, 1024] @ W[1024, 1024] + bias[1024]
// Block: 256 threads (8 waves). Block tile: 128(M) x 64(N).
// Wave w computes rows [wtile, wtile+16) x cols [nbase, nbase+64) with 4
// v8f accumulators; one shared fp32 A-fragment feeds 4 WMMAs per k-step.
// ---------------------------------------------------------------------------
__global__ __launch_bounds__(256) void gemm_bias_wmma(
    const float* __restrict__ A, const float* __restrict__ W,
    const float* __restrict__ bias, float* __restrict__ C)
{
  constexpr int N = DM, K = DM;
  const int lane  = threadIdx.x & 31;
  const int wave  = threadIdx.x >> 5;
  const int ln    = lane & 15;   // M (A) / N (B,C) index within fragment
  const int lh    = lane >> 4;   // selects K pair {0,1} vs {2,3}
  const int nbase = blockIdx.x * 64;
  const int mtile = blockIdx.y * 128 + wave * 16;

  const float* Arow = A + (size_t)(mtile + ln) * K + 2 * lh;
  const float* Wcol = W + (size_t)(2 * lh) * N + nbase + ln;

  v8f acc0 = {}, acc1 = {}, acc2 = {}, acc3 = {};
#pragma unroll 4
  for (int k0 = 0; k0 < K; k0 += 4) {
    v2f a = *(const v2f*)(Arow + k0);           // A(m, k0+2lh .. +1)
    const float* w0 = Wcol + (size_t)k0 * N;    // W row k0+2lh
    const float* w1 = w0 + N;                   // W row k0+2lh+1
    v2f b;
    b.x = w0[0];  b.y = w1[0];  acc0 = WMMA_F32(a, b, acc0);
    b.x = w0[16]; b.y = w1[16]; acc1 = WMMA_F32(a, b, acc1);
    b.x = w0[32]; b.y = w1[32]; acc2 = WMMA_F32(a, b, acc2);
    b.x = w0[48]; b.y = w1[48]; acc3 = WMMA_F32(a, b, acc3);
  }

  const float b0 = bias[nbase + ln];
  const float b1 = bias[nbase + 16 + ln];
  const float b2 = bias[nbase + 32 + ln];
  const float b3 = bias[nbase + 48 + ln];
#pragma unroll
  for (int i = 0; i < 8; ++i) {
    const int r = mtile + i + 8 * lh;           // C/D row mapping
    float* Crow = C + (size_t)r * N + nbase + ln;
    Crow[0]  = acc0[i] + b0;
    Crow[16] = acc1[i] + b1;
    Crow[32] = acc2[i] + b2;
    Crow[48] = acc3[i] + b3;
  }
}

// ---------------------------------------------------------------------------
// Causal flash attention, fp32 WMMA.
// Q,K,V stored [B, S, NH*DH]; output same layout.
// Grid: (S/128, NH, B). Block: 256 threads = 8 waves.
// Wave w owns query rows [qbase, qbase+16) of head (b,h); loops causal key
// tiles of 16 with online softmax. All synchronization is intra-wave.
// ---------------------------------------------------------------------------
__global__ __launch_bounds__(256) void attn_causal_wmma(
    const float* __restrict__ Q, const float* __restrict__ Km,
    const float* __restrict__ V, float* __restrict__ O)
{
  __shared__ float plds[8 * 16 * 18];           // per-wave 16x16 P tile, pad 18
  const int lane  = threadIdx.x & 31;
  const int wave  = threadIdx.x >> 5;
  const int ln    = lane & 15;
  const int lh    = lane >> 4;
  const int b     = blockIdx.z;
  const int h     = blockIdx.y;
  const int qbase = blockIdx.x * 128 + wave * 16;

  const size_t base = (size_t)b * SEQ * DM + (size_t)h * DH;
  const float* Qp = Q  + base;
  const float* Kp = Km + base;
  const float* Vp = V  + base;

  // Register-resident Q fragments: 16 k-steps covering d_head=64.
  v2f qa[16];
#pragma unroll
  for (int t = 0; t < 16; ++t)
    qa[t] = *(const v2f*)(Qp + (size_t)(qbase + ln) * DM + t * 4 + 2 * lh);

  v8f o0 = {}, o1 = {}, o2 = {}, o3 = {};
  float mi[8], li[8];
#pragma unroll
  for (int i = 0; i < 8; ++i) { mi[i] = -__builtin_inff(); li[i] = 0.0f; }

  float* pw = plds + wave * (16 * 18);
  const float scale = 0.125f;                   // 1/sqrt(64)

  const int ktiles = qbase / 16 + 1;            // causal: keys up to qbase+15
  for (int kt0 = 0; kt0 < ktiles; ++kt0) {
    const int kt = kt0 * 16;

    // ---- S = Q @ K^T for this 16x16 tile (K-dim = 64, 16 WMMAs) ----
    v8f s = {};
    const float* Kcol = Kp + (size_t)(kt + ln) * DM + 2 * lh;  // B(k,n)=K[n][k]
#pragma unroll
    for (int t = 0; t < 16; ++t) {
      v2f bf = *(const v2f*)(Kcol + t * 4);
      s = WMMA_F32(qa[t], bf, s);
    }

    // ---- online softmax (row = i + 8*lh lives in one 16-lane half) ----
#pragma unroll
    for (int i = 0; i < 8; ++i) {
      float sv = s[i] * scale;
      const int qrow = qbase + i + 8 * lh;
      const int kcol = kt + ln;
      if (kcol > qrow) sv = -__builtin_inff();
      float rmax = sv;
      rmax = fmaxf(rmax, __shfl_xor(rmax, 1, 16));
      rmax = fmaxf(rmax, __shfl_xor(rmax, 2, 16));
      rmax = fmaxf(rmax, __shfl_xor(rmax, 4, 16));
      rmax = fmaxf(rmax, __shfl_xor(rmax, 8, 16));
      const float mnew = fmaxf(mi[i], rmax);
      const float p    = __expf(sv - mnew);
      const float corr = __expf(mi[i] - mnew);  // 0 on first tile (mi=-inf)
      float rsum = p;
      rsum += __shfl_xor(rsum, 1, 16);
      rsum += __shfl_xor(rsum, 2, 16);
      rsum += __shfl_xor(rsum, 4, 16);
      rsum += __shfl_xor(rsum, 8, 16);
      li[i] = li[i] * corr + rsum;
      mi[i] = mnew;
      s[i]  = p;
      o0[i] *= corr; o1[i] *= corr; o2[i] *= corr; o3[i] *= corr;
    }

    // ---- C-layout -> A-layout conversion through LDS (intra-wave) ----
#pragma unroll
    for (int i = 0; i < 8; ++i)
      pw[(i + 8 * lh) * 18 + ln] = s[i];
    asm volatile("s_wait_dscnt 0" ::: "memory"); // stores visible to own wave

    // ---- O += P @ V  (K-dim = 16 keys, 16 WMMAs) ----
#pragma unroll
    for (int t2 = 0; t2 < 4; ++t2) {
      v2f pa = *(const v2f*)(pw + ln * 18 + t2 * 4 + 2 * lh);  // A(m,k) frag
      const float* Vr = Vp + (size_t)(kt + t2 * 4 + 2 * lh) * DM + ln;
      v2f bf;
      bf.x = Vr[0];  bf.y = Vr[DM];      o0 = WMMA_F32(pa, bf, o0);
      bf.x = Vr[16]; bf.y = Vr[DM + 16]; o1 = WMMA_F32(pa, bf, o1);
      bf.x = Vr[32]; bf.y = Vr[DM + 32]; o2 = WMMA_F32(pa, bf, o2);
      bf.x = Vr[48]; bf.y = Vr[DM + 48]; o3 = WMMA_F32(pa, bf, o3);
    }
    asm volatile("" ::: "memory");  // keep next tile's LDS stores after loads
  }

  // ---- epilogue: normalize and scatter back to [B,S,NH*DH] ----
#pragma unroll
  for (int i = 0; i < 8; ++i) {
    const float inv = 1.0f / li[i];
    const int r = qbase + i + 8 * lh;
    float* Orow = O + ((size_t)b * SEQ + r) * DM + (size_t)h * DH + ln;
    Orow[0]  = o0[i] * inv;
    Orow[16] = o1[i] * inv;
    Orow[32] = o2[i] * inv;
    Orow[48] = o3[i] * inv;
  }
}

// ---------------------------------------------------------------------------
extern "C" void kernel_launch(void* const* d_in, const int* in_sizes, int n_in,
                              void* d_out, int out_size, void* d_ws, size_t ws_size,
                              hipStream_t stream) {
  (void)in_sizes; (void)n_in; (void)out_size; (void)ws_size;
  const float* x  = (const float*)d_in[0];
  const float* Wq = (const float*)d_in[1];
  const float* bq = (const float*)d_in[2];
  const float* Wk = (const float*)d_in[3];
  const float* bk = (const float*)d_in[4];
  const float* Wv = (const float*)d_in[5];
  const float* bv = (const float*)d_in[6];
  const float* Wo = (const float*)d_in[7];
  const float* bo = (const float*)d_in[8];
  float* out = (float*)d_out;

  const size_t plane = (size_t)MROWS * DM;      // 4096*1024 floats = 16 MB
  float* Qb = (float*)d_ws;
  float* Kb = Qb + plane;
  float* Vb = Kb + plane;
  float* Ab = Vb + plane;                        // attention output, pre-proj

  const dim3 gblk(256);
  const dim3 ggrid(DM / 64, MROWS / 128);        // (16, 32)
  gemm_bias_wmma<<<ggrid, gblk, 0, stream>>>(x, Wq, bq, Qb);
  gemm_bias_wmma<<<ggrid, gblk, 0, stream>>>(x, Wk, bk, Kb);
  gemm_bias_wmma<<<ggrid, gblk, 0, stream>>>(x, Wv, bv, Vb);

  const dim3 agrid(SEQ / 128, NH, BATCH);        // (16, 16, 2)
  attn_causal_wmma<<<agrid, gblk, 0, stream>>>(Qb, Kb, Vb, Ab);

  gemm_bias_wmma<<<ggrid, gblk, 0, stream>>>(Ab, Wo, bo, out);
}